// Experts_51462298141169
// MI455X (gfx1250) — compile-verified
//
#include <hip/hip_runtime.h>

typedef __attribute__((ext_vector_type(2))) float v2f;
typedef __attribute__((ext_vector_type(8))) float v8f;

#define HDIM   1024
#define DFF    4096
#define NEXP   4
#define CAP    2048
#define TTILE  16        // tokens per workgroup
#define NWAVE  8         // waves per workgroup
#define FCHUNK 128       // DFF chunk = NWAVE*16
#define HSTR   132       // padded LDS row stride (floats) to avoid bank conflicts

// Branch-free tanh: prefer the gfx1250 hardware V_TANH; otherwise use
// v_exp_f32 + v_rcp_f32 (saturates correctly for |y| large, no EXEC divergence).
__device__ __forceinline__ float fast_tanh(float y) {
#if __has_builtin(__builtin_amdgcn_tanhf)
    return __builtin_amdgcn_tanhf(y);
#else
    float e = __builtin_amdgcn_exp2f(2.8853900817779268f * y);  // exp(2y)
    return 1.0f - 2.0f * __builtin_amdgcn_rcpf(e + 1.0f);
#endif
}

__device__ __forceinline__ float gelu_tanh(float x) {
    // jax.nn.gelu default (approximate=True)
    float x3 = x * x * x;
    float t  = fast_tanh(0.7978845608028654f * (x + 0.044715f * x3));
    return 0.5f * x * (1.0f + t);
}

__global__ __launch_bounds__(256) void Experts_51462298141169_kernel(
    const float* __restrict__ X,    // [W,E,C,H]
    const float* __restrict__ W1,   // [E,H,DFF]
    const float* __restrict__ B1,   // [E,DFF]
    const float* __restrict__ W2,   // [E,DFF,H]
    const float* __restrict__ B2,   // [E,H]
    float* __restrict__ Out)        // [W,E,C,H]
{
    __shared__ float hbuf[TTILE * HSTR];   // 8448 B

    const int tid  = threadIdx.x;
    const int lane = tid & 31;
    const int wave = tid >> 5;        // 0..7
    const int half = lane >> 4;       // 0/1 (half-wave)
    const int ln   = lane & 15;       // 0..15

    const long tokbase = (long)blockIdx.x * TTILE;          // 16 tokens, never crosses expert
    const int  e       = (int)((tokbase / CAP) & (NEXP - 1));

    const float* Xb  = X  + (size_t)tokbase * HDIM;
    const float* W1e = W1 + (size_t)e * HDIM * DFF;
    const float* B1e = B1 + (size_t)e * DFF;
    const float* W2e = W2 + (size_t)e * DFF * HDIM;
    const float* B2e = B2 + (size_t)e * HDIM;
    float*       Ob  = Out + (size_t)tokbase * HDIM;

    // Output accumulators: 8 tiles of 16x16 covering columns [wave*128, wave*128+128)
    v8f acc[8] = {};
    const int jbase = wave * 128;

    // A operand (GEMM1): row m = ln, K pair offset 2*half  -> b64 loads
    const float* Arow = Xb + (size_t)ln * HDIM + 2 * half;

    for (int fc = 0; fc < DFF; fc += FCHUNK) {
        // ---------- GEMM1: this wave computes h[:, fc + wave*16 .. +16) ----------
        const int fcol = fc + wave * 16;
        const float* Bcol = W1e + (size_t)(2 * half) * DFF + fcol + ln;

        v8f s0 = {}, s1 = {};
        #pragma unroll 4
        for (int k = 0; k < HDIM; k += 8) {
            v2f a0 = *(const v2f*)(Arow + k);
            v2f b0;
            b0.x = Bcol[(size_t)(k + 0) * DFF];
            b0.y = Bcol[(size_t)(k + 1) * DFF];
            s0 = __builtin_amdgcn_wmma_f32_16x16x4_f32(false, a0, false, b0,
                                                       (short)0, s0, false, false);
            v2f a1 = *(const v2f*)(Arow + k + 4);
            v2f b1v;
            b1v.x = Bcol[(size_t)(k + 4) * DFF];
            b1v.y = Bcol[(size_t)(k + 5) * DFF];
            s1 = __builtin_amdgcn_wmma_f32_16x16x4_f32(false, a1, false, b1v,
                                                       (short)0, s1, false, false);
        }
        v8f s = s0 + s1;

        // bias + gelu + stage into LDS (row m = i + 8*half, col = wave*16 + ln)
        const float bias1 = B1e[fcol + ln];
        {
            float* hp = hbuf + (size_t)(8 * half) * HSTR + wave * 16 + ln;
            #pragma unroll
            for (int i = 0; i < 8; ++i)
                hp[i * HSTR] = gelu_tanh(s[i] + bias1);
        }
        __syncthreads();

        // ---------- GEMM2: acc[:, jbase..jbase+128) += h_chunk * W2_chunk ----------
        const float* hA     = hbuf + (size_t)ln * HSTR + 2 * half;
        const float* W2base = W2e + (size_t)(fc + 2 * half) * HDIM + jbase + ln;

        #pragma unroll 2
        for (int kc = 0; kc < FCHUNK; kc += 4) {
            v2f a = *(const v2f*)(hA + kc);          // ds_load_b64
            const float* Bp = W2base + (size_t)kc * HDIM;
            #pragma unroll
            for (int t = 0; t < 8; ++t) {
                v2f b;
                b.x = Bp[t * 16];
                b.y = Bp[(size_t)HDIM + t * 16];
                acc[t] = __builtin_amdgcn_wmma_f32_16x16x4_f32(false, a, false, b,
                                                               (short)0, acc[t],
                                                               false, false);
            }
        }
        __syncthreads();   // before overwriting hbuf next chunk
    }

    // ---------- epilogue: + b2, store [16 x 128] per wave ----------
    #pragma unroll
    for (int t = 0; t < 8; ++t) {
        const int   j     = jbase + t * 16 + ln;
        const float bias2 = B2e[j];
        float* op = Ob + (size_t)(8 * half) * HDIM + j;
        #pragma unroll
        for (int i = 0; i < 8; ++i)
            op[(size_t)i * HDIM] = acc[t][i] + bias2;
    }
}

extern "C" void kernel_launch(void* const* d_in, const int* in_sizes, int n_in,
                              void* d_out, int out_size, void* d_ws, size_t ws_size,
                              hipStream_t stream) {
    (void)n_in; (void)d_ws; (void)ws_size; (void)out_size;
    const float* X  = (const float*)d_in[0];
    const float* W1 = (const float*)d_in[1];
    const float* B1 = (const float*)d_in[2];
    const float* W2 = (const float*)d_in[3];
    const float* B2 = (const float*)d_in[4];
    float* Out = (float*)d_out;

    const int tokens = in_sizes[0] / HDIM;        // W*E*C = 65536
    const int grid   = tokens / TTILE;            // 4096 workgroups
    Experts_51462298141169_kernel<<<grid, 256, 0, stream>>>(X, W1, B1, W2, B2, Out);
}